// GCN_31018253812301
// MI455X (gfx1250) — compile-verified
//
#include <hip/hip_runtime.h>
#include <hip/hip_bf16.h>
#include <math.h>

typedef float v2f __attribute__((ext_vector_type(2)));
typedef float v8f __attribute__((ext_vector_type(8)));
typedef int   gv4i __attribute__((vector_size(4 * sizeof(int))));  // matches builtin

#if defined(__has_builtin)
#if __has_builtin(__builtin_amdgcn_wmma_f32_16x16x4_f32)
#define HAVE_WMMA_F32 1
#endif
#if __has_builtin(__builtin_amdgcn_global_load_async_to_lds_b128) && \
    __has_builtin(__builtin_amdgcn_global_load_async_to_lds_b32)
#define HAVE_ASYNC 1
#endif
#endif

#define NEG_INF (-1.0e12f)

// ---------------------------------------------------------------------------
// async global -> LDS helpers (CDNA5 GLOBAL_LOAD_ASYNC_TO_LDS_*, ASYNCcnt)
// builtin signature (from clang diagnostic): arg0 = T __attribute__((
// vector_size)) * in addrspace(1) (non-const), arg1 = LDS dst, then 2 imms.
// ---------------------------------------------------------------------------
__device__ __forceinline__ void async_cp_b128(const float* gsrc, float* ldst)
{
#if HAVE_ASYNC
    float* gs = const_cast<float*>(gsrc);
    __builtin_amdgcn_global_load_async_to_lds_b128(
        (__attribute__((address_space(1))) gv4i*)gs,
        (__attribute__((address_space(3))) gv4i*)ldst, 0, 0);
#else
    *(float4*)ldst = *(const float4*)gsrc;
#endif
}

__device__ __forceinline__ void async_cp_b32(const float* gsrc, float* ldst)
{
#if HAVE_ASYNC
    float* gs = const_cast<float*>(gsrc);
    __builtin_amdgcn_global_load_async_to_lds_b32(
        (__attribute__((address_space(1))) int*)gs,
        (__attribute__((address_space(3))) int*)ldst, 0, 0);
#else
    *ldst = *gsrc;
#endif
}

__device__ __forceinline__ void wait_async()
{
#if HAVE_ASYNC
#if __has_builtin(__builtin_amdgcn_s_wait_asynccnt)
    __builtin_amdgcn_s_wait_asynccnt(0);
#else
    asm volatile("s_wait_asynccnt 0x0" ::: "memory");
#endif
#endif
}

// ---------------------------------------------------------------------------
// Generic batched GEMM: C = alpha * A @ B (+ bias), fp32, WMMA f32 16x16x4.
//   A: [M x K] row-major, lda
//   B: bNK==1 -> stored [N x K] row-major (B_mem[n][k])
//      bNK==0 -> stored [K x N] row-major (B_mem[k][n])
//   C: [M x N] row-major, ldc. blockIdx.z = batch.
// Tile: 128x128 per block, 8 waves, each wave 64x32 (4x2 WMMA 16x16 tiles).
// Double-buffered LDS, async global->LDS staging.
// Requires M%128==0, N%128==0, K%32==0 (true for all launches here).
// ---------------------------------------------------------------------------
#define BM 128
#define BN 128
#define BK 32
#define LDSP (BK + 4)   // pad 4: 16B-aligned b128 LDS dst, 8B-aligned v2f loads

__device__ __forceinline__ void stage_tile(
    float (*sAb)[LDSP], float (*sBb)[LDSP],
    const float* __restrict__ A, int lda,
    const float* __restrict__ B, int ldb,
    int blockM0, int blockN0, int kt, int tid, int bNK)
{
    for (int i = 0; i < 4; ++i) {
        int idx = tid + i * 256;            // 1024 float4 slots: 128 x 8
        int r = idx >> 3;
        int c = (idx & 7) << 2;
        async_cp_b128(&A[(long)(blockM0 + r) * lda + kt + c], &sAb[r][c]);
    }
    if (bNK) {
        for (int i = 0; i < 4; ++i) {
            int idx = tid + i * 256;
            int n = idx >> 3;
            int c = (idx & 7) << 2;
            async_cp_b128(&B[(long)(blockN0 + n) * ldb + kt + c], &sBb[n][c]);
        }
    } else {
        // transpose [K x N] -> LDS [N][K]; consecutive tids read consecutive n
        for (int i = 0; i < 16; ++i) {
            int idx = tid + i * 256;        // 4096 scalars: 32 k x 128 n
            int n = idx & 127;
            int k = idx >> 7;
            async_cp_b32(&B[(long)(kt + k) * ldb + blockN0 + n], &sBb[n][k]);
        }
    }
}

__global__ __launch_bounds__(256) void gemm_f32_wmma(
    const float* __restrict__ A, int lda, long strideA,
    const float* __restrict__ B, int ldb, long strideB,
    float* __restrict__ C, int ldc, long strideC,
    const float* __restrict__ bias,
    int K, float alpha, int bNK)
{
    __shared__ __align__(16) float sA[2][BM][LDSP];
    __shared__ __align__(16) float sB[2][BN][LDSP];

    const int tid = threadIdx.x;
    const long batch = blockIdx.z;
    A += batch * strideA;
    B += batch * strideB;
    C += batch * strideC;

    const int blockN0 = blockIdx.x * BN;
    const int blockM0 = blockIdx.y * BM;

#if HAVE_WMMA_F32
    const int lid = tid & 31;
    const int wid = tid >> 5;
    const int waveM = (wid & 1) * 64;   // 2 waves along M
    const int waveN = (wid >> 1) * 32;  // 4 waves along N
    const int l16 = lid & 15;
    const int half = lid >> 4;          // lanes 16-31: K+2..3 (frag), M+8..15 (D)

    v8f acc[4][2];
    for (int mt = 0; mt < 4; ++mt)
        for (int nt = 0; nt < 2; ++nt)
            for (int e = 0; e < 8; ++e) acc[mt][nt][e] = 0.0f;
#else
    const int tx = tid & 15;            // fallback: 8x8 per-thread tile
    const int ty = tid >> 4;
    float creg[8][8];
    for (int i = 0; i < 8; ++i)
        for (int j = 0; j < 8; ++j) creg[i][j] = 0.0f;
#endif

    // prologue: stage first K-chunk into buffer 0
    stage_tile(sA[0], sB[0], A, lda, B, ldb, blockM0, blockN0, 0, tid, bNK);

    int buf = 0;
    for (int kt = 0; kt < K; kt += BK) {
        wait_async();        // own async copies into `buf` complete
        __syncthreads();     // all waves' copies complete; readers of buf^1 done

        if (kt + BK < K)     // stream next chunk while computing this one
            stage_tile(sA[buf ^ 1], sB[buf ^ 1], A, lda, B, ldb,
                       blockM0, blockN0, kt + BK, tid, bNK);

#if HAVE_WMMA_F32
        for (int ks = 0; ks < BK; ks += 4) {
            const int kb = ks + (half << 1);   // lanes<16: K+0..1, >=16: K+2..3
            v2f af[4], bf[2];
            for (int mt = 0; mt < 4; ++mt)
                af[mt] = *(const v2f*)&sA[buf][waveM + mt * 16 + l16][kb];
            for (int nt = 0; nt < 2; ++nt)
                bf[nt] = *(const v2f*)&sB[buf][waveN + nt * 16 + l16][kb];
            for (int mt = 0; mt < 4; ++mt)
                for (int nt = 0; nt < 2; ++nt)
                    acc[mt][nt] = __builtin_amdgcn_wmma_f32_16x16x4_f32(
                        false, af[mt], false, bf[nt],
                        (short)0, acc[mt][nt], false, false);
        }
#else
        for (int k = 0; k < BK; ++k) {
            float a[8], b[8];
            for (int i = 0; i < 8; ++i) a[i] = sA[buf][ty * 8 + i][k];
            for (int j = 0; j < 8; ++j) b[j] = sB[buf][tx * 8 + j][k];
            for (int i = 0; i < 8; ++i)
                for (int j = 0; j < 8; ++j)
                    creg[i][j] = fmaf(a[i], b[j], creg[i][j]);
        }
#endif
        buf ^= 1;
    }

    // ---- epilogue: D layout (f32 16x16): lane<16 -> M=r, lane>=16 -> M=r+8 ----
#if HAVE_WMMA_F32
    for (int mt = 0; mt < 4; ++mt)
        for (int nt = 0; nt < 2; ++nt) {
            const int col = blockN0 + waveN + nt * 16 + l16;
            const float bv = bias ? bias[col] : 0.0f;
            const int rbase = blockM0 + waveM + mt * 16 + (half << 3);
            for (int r = 0; r < 8; ++r)
                C[(long)(rbase + r) * ldc + col] = alpha * acc[mt][nt][r] + bv;
        }
#else
    for (int j = 0; j < 8; ++j) {
        const int col = blockN0 + tx * 8 + j;
        const float bv = bias ? bias[col] : 0.0f;
        for (int i = 0; i < 8; ++i)
            C[(long)(blockM0 + ty * 8 + i) * ldc + col] = alpha * creg[i][j] + bv;
    }
#endif
}

// ---------------------------------------------------------------------------
// Row softmax over 512 columns, one block per row (src may equal dst).
// ---------------------------------------------------------------------------
__global__ __launch_bounds__(256) void softmax512(const float* __restrict__ src,
                                                  float* __restrict__ dst)
{
    const long row = blockIdx.x;
    const float* s = src + row * 512;
    float* d = dst + row * 512;
    const int tid = threadIdx.x;
    float v0 = s[tid], v1 = s[tid + 256];

    __shared__ float red[256];
    red[tid] = fmaxf(v0, v1);
    __syncthreads();
    for (int off = 128; off > 0; off >>= 1) {
        if (tid < off) red[tid] = fmaxf(red[tid], red[tid + off]);
        __syncthreads();
    }
    const float m = red[0];
    __syncthreads();

    const float e0 = __expf(v0 - m), e1 = __expf(v1 - m);
    red[tid] = e0 + e1;
    __syncthreads();
    for (int off = 128; off > 0; off >>= 1) {
        if (tid < off) red[tid] += red[tid + off];
        __syncthreads();
    }
    const float inv = 1.0f / red[0];
    d[tid] = e0 * inv;
    d[tid + 256] = e1 * inv;
}

// ---------------------------------------------------------------------------
// node mask: mask[b][s] = (sum_t adj[b][s][t] + sum_t adj[b][t][s]) == 0
// ---------------------------------------------------------------------------
__global__ __launch_bounds__(256) void node_mask_k(const float* __restrict__ adj,
                                                   float* __restrict__ om)
{
    const int bs = blockIdx.x;
    const int b = bs >> 9, s = bs & 511;
    const float* A = adj + (long)b * 512 * 512;
    const int tid = threadIdx.x;
    float sum = 0.0f;
    for (int t = tid; t < 512; t += 256)
        sum += A[(long)s * 512 + t] + A[(long)t * 512 + s];
    __shared__ float red[256];
    red[tid] = sum;
    __syncthreads();
    for (int off = 128; off > 0; off >>= 1) {
        if (tid < off) red[tid] += red[tid + off];
        __syncthreads();
    }
    if (tid == 0) om[bs] = (red[0] == 0.0f) ? 1.0f : 0.0f;
}

// ---------------------------------------------------------------------------
// token embeddings: masked max-pool over W=8 word pieces (no -INF fixup)
// ---------------------------------------------------------------------------
__global__ __launch_bounds__(256) void pool_g_k(const int* __restrict__ words,
                                                const int* __restrict__ masks,
                                                const float* __restrict__ emb,
                                                float* __restrict__ g)
{
    const int bs = blockIdx.x;
    const int base = bs * 8;
    for (int d = threadIdx.x; d < 768; d += 256) {
        float m = NEG_INF;
        for (int w = 0; w < 8; ++w) {
            float v = masks[base + w] ? NEG_INF
                                      : emb[(long)words[base + w] * 768 + d];
            m = fmaxf(m, v);
        }
        g[(long)bs * 768 + d] = m;
    }
}

// main: masked max-pool over Wm=64, then -INF -> 0
__global__ __launch_bounds__(256) void pool_main_k(const int* __restrict__ mw,
                                                   const int* __restrict__ mm,
                                                   const float* __restrict__ emb,
                                                   float* __restrict__ o)
{
    const int bm = blockIdx.x;
    const int base = bm * 64;
    for (int d = threadIdx.x; d < 768; d += 256) {
        float mx = NEG_INF;
        for (int w = 0; w < 64; ++w) {
            float v = mm[base + w] ? NEG_INF
                                   : emb[(long)mw[base + w] * 768 + d];
            mx = fmaxf(mx, v);
        }
        if (mx == NEG_INF) mx = 0.0f;
        o[(long)bm * 768 + d] = mx;
    }
}

// user: masked max-pool over Wu=16 then max over U=32 users, then -INF -> 0
__global__ __launch_bounds__(256) void pool_user_k(const int* __restrict__ uw,
                                                   const int* __restrict__ um,
                                                   const float* __restrict__ emb,
                                                   float* __restrict__ o)
{
    const int b = blockIdx.x;
    for (int d = threadIdx.x; d < 768; d += 256) {
        float best = NEG_INF;
        for (int u = 0; u < 32; ++u) {
            const int base = (b * 32 + u) * 16;
            float mx = NEG_INF;
            for (int w = 0; w < 16; ++w) {
                float v = um[base + w] ? NEG_INF
                                       : emb[(long)uw[base + w] * 768 + d];
                mx = fmaxf(mx, v);
            }
            best = fmaxf(best, mx);
        }
        if (best == NEG_INF) best = 0.0f;
        o[(long)b * 768 + d] = best;
    }
}

// ---------------------------------------------------------------------------
extern "C" void kernel_launch(void* const* d_in, const int* in_sizes, int n_in,
                              void* d_out, int out_size, void* d_ws, size_t ws_size,
                              hipStream_t stream)
{
    (void)in_sizes; (void)n_in; (void)out_size; (void)ws_size;
    const int B = 16, S = 512, D = 768, M_ = 2;

    const int*   words     = (const int*)d_in[0];
    const int*   masks     = (const int*)d_in[1];
    const int*   main_w    = (const int*)d_in[2];
    const int*   main_mask = (const int*)d_in[3];
    const int*   user_w    = (const int*)d_in[4];
    const int*   user_mask = (const int*)d_in[5];
    const float* adj       = (const float*)d_in[6];
    const float* emb       = (const float*)d_in[7];
    const float* K_w       = (const float*)d_in[8];
    const float* K_b       = (const float*)d_in[9];
    const float* Q_w       = (const float*)d_in[10];
    const float* Q_b       = (const float*)d_in[11];

    float* out    = (float*)d_out;
    float* o_gcn  = out;                          // [B,S,D]
    float* o_mask = o_gcn  + (long)B * S * D;     // [B,S]
    float* o_dg   = o_mask + (long)B * S;         // [B,S,S]
    float* o_adj  = o_dg   + (long)B * S * S;     // [B,S,S]
    float* o_main = o_adj  + (long)B * S * S;     // [B,M,D]
    float* o_user = o_main + (long)B * M_ * D;    // [B,D]

    float* g   = (float*)d_ws;                    // [B,S,D]
    float* key = g   + (long)B * S * D;           // [B,S,D]
    float* qry = key + (long)B * S * D;           // [B,S,D]

    const long SD = (long)S * D;                  // 393216
    const long SS = (long)S * S;                  // 262144
    const float inv_sqrt_d = 1.0f / sqrtf((float)D);

    // pooled embeddings + cheap outputs
    pool_g_k   <<<B * S,  256, 0, stream>>>(words, masks, emb, g);
    pool_main_k<<<B * M_, 256, 0, stream>>>(main_w, main_mask, emb, o_main);
    pool_user_k<<<B,      256, 0, stream>>>(user_w, user_mask, emb, o_user);
    node_mask_k<<<B * S,  256, 0, stream>>>(adj, o_mask);
    softmax512 <<<B * S,  256, 0, stream>>>(adj, o_adj);

    // key = g @ K_w^T + K_b ; query = g @ Q_w^T + Q_b   (M=8192, N=768, K=768)
    {
        dim3 grid(D / BN, (B * S) / BM, 1);
        gemm_f32_wmma<<<grid, 256, 0, stream>>>(g, D, 0, K_w, D, 0,
                                                key, D, 0, K_b, D, 1.0f, 1);
        gemm_f32_wmma<<<grid, 256, 0, stream>>>(g, D, 0, Q_w, D, 0,
                                                qry, D, 0, Q_b, D, 1.0f, 1);
    }

    // scores[b] = key[b] @ qry[b]^T / sqrt(D)   (M=512, N=512, K=768, batched)
    {
        dim3 grid(S / BN, S / BM, B);
        gemm_f32_wmma<<<grid, 256, 0, stream>>>(key, D, SD, qry, D, SD,
                                                o_dg, S, SS, nullptr,
                                                D, inv_sqrt_d, 1);
    }

    // dense_graph = softmax(scores, axis=2) in place
    softmax512<<<B * S, 256, 0, stream>>>(o_dg, o_dg);

    // gcn_out[b] = dense_graph[b] @ g[b]   (M=512, N=768, K=512, batched)
    {
        dim3 grid(D / BN, S / BM, B);
        gemm_f32_wmma<<<grid, 256, 0, stream>>>(o_dg, S, SS, g, D, SD,
                                                o_gcn, D, SD, nullptr,
                                                S, 1.0f, 0);
    }
}